// MSEBPRLoss_65532611002862
// MI455X (gfx1250) — compile-verified
//
#include <hip/hip_runtime.h>
#include <stdint.h>

#ifndef __has_builtin
#define __has_builtin(x) 0
#endif

#define N_ELEMS 16384
#define BLOCK   256
#define IPT     4
#define ITILE   (BLOCK * IPT)        // 1024 i per block
#define GI      (N_ELEMS / ITILE)    // 16
#define JCHUNK  512
#define GJ      (N_ELEMS / JCHUNK)   // 32
#define STAGE   256                  // j per LDS stage
#define NSTAGES (JCHUNK / STAGE)     // 2 (double-buffered)
#define NBLOCKS (GI * GJ)            // 512 partial sums

// ---------------- CDNA5 async global->LDS helpers ----------------

typedef int v2i __attribute__((ext_vector_type(2)));
typedef __attribute__((address_space(1))) v2i g_v2i;
typedef __attribute__((address_space(3))) v2i l_v2i;

__device__ __forceinline__ void wait_async_le1() {
#if __has_builtin(__builtin_amdgcn_s_wait_asynccnt)
  __builtin_amdgcn_s_wait_asynccnt(1);
#else
  asm volatile("s_wait_asynccnt 1" ::: "memory");
#endif
}

__device__ __forceinline__ void wait_async_le0() {
#if __has_builtin(__builtin_amdgcn_s_wait_asynccnt)
  __builtin_amdgcn_s_wait_asynccnt(0);
#else
  asm volatile("s_wait_asynccnt 0" ::: "memory");
#endif
}

// Copy 8 bytes (2 floats) per lane from global to LDS asynchronously.
__device__ __forceinline__ void async_load_f2(const float* gsrc, float* ldst) {
#if __has_builtin(__builtin_amdgcn_global_load_async_to_lds_b64)
  __builtin_amdgcn_global_load_async_to_lds_b64(
      (g_v2i*)(uintptr_t)gsrc,   // global AS1 (flat==global address values)
      (l_v2i*)ldst,              // true addrspacecast generic -> LDS
      0, 0);
#else
  // Synchronous fallback: ASYNCcnt stays 0, waits are no-ops.
  float2 v = *(const float2*)gsrc;
  *(float2*)ldst = v;
#endif
}

// ---------------- main pairwise kernel ----------------

__global__ __launch_bounds__(BLOCK) void
mse_bpr_pair_kernel(const float* __restrict__ x, const float* __restrict__ t,
                    float* __restrict__ ws) {
  __shared__ float sx[2][STAGE];
  __shared__ float st[2][STAGE];
  __shared__ float red[BLOCK];

  const int tid = threadIdx.x;
  const int ib  = blockIdx.x;          // i-tile index, 0..GI-1
  const int jb  = blockIdx.y;          // j-chunk index, 0..GJ-1
  const int j0  = jb * JCHUNK;

  // Per-thread i values (strided by BLOCK for coalesced loads).
  // Fused-mse precompute: mse_i = (x_i - t_i)^2, ximod_i = x_i + mse_i.
  // Masked pair contribution (raw):
  //   sp + mse_i = ln2*log2(1+exp2(-log2e*|d|)) + max(d,0) + mse_i
  //              = fma(log2(1+exp2(-log2e*|d|)), ln2, max(ximod_i - x_j, mse_i))
  float xi[IPT], ti[IPT], ximod[IPT], msei[IPT];
#pragma unroll
  for (int k = 0; k < IPT; ++k) {
    const int i = ib * ITILE + k * BLOCK + tid;
    xi[k]    = x[i];
    ti[k]    = t[i];
    const float dm = xi[k] - ti[k];
    msei[k]  = dm * dm;
    ximod[k] = xi[k] + msei[k];
  }
  float acc0 = 0.0f, acc1 = 0.0f;

  // Prologue: stage 0 async. Waves 0-3 (tid<128) stage x, waves 4-7 stage t,
  // so every wave issues exactly one async instruction per stage.
  if (tid < 128) async_load_f2(&x[j0 + 2 * tid],         &sx[0][2 * tid]);
  else           async_load_f2(&t[j0 + 2 * (tid - 128)], &st[0][2 * (tid - 128)]);

  for (int s = 0; s < NSTAGES; ++s) {
    const int buf = s & 1;
    if (s + 1 < NSTAGES) {
      const int nb = (s + 1) & 1;
      const int jn = j0 + (s + 1) * STAGE;
      if (tid < 128) async_load_f2(&x[jn + 2 * tid],         &sx[nb][2 * tid]);
      else           async_load_f2(&t[jn + 2 * (tid - 128)], &st[nb][2 * (tid - 128)]);
      wait_async_le1();   // newest stage still in flight; stage s complete
    } else {
      wait_async_le0();
    }
    __syncthreads();      // make completed async LDS writes visible to all waves

#pragma unroll 4
    for (int jj = 0; jj < STAGE; ++jj) {
      const float xj = sx[buf][jj];
      const float tj = st[buf][jj];
#pragma unroll
      for (int k = 0; k < IPT; ++k) {
        const float d   = xi[k] - xj;
        const float e   = __builtin_amdgcn_exp2f(-1.4426950408889634f *
                                                 __builtin_fabsf(d));
        const float mx2 = __builtin_fmaxf(ximod[k] - xj, msei[k]);
        const float v   = __builtin_fmaf(__builtin_amdgcn_logf(1.0f + e),
                                         0.6931471805599453f, mx2);
        const float sel = (tj > ti[k]) ? v : 0.0f;   // v_cmp + v_cndmask
        if (k & 1) acc1 += sel; else acc0 += sel;    // 2 chains
      }
    }
    __syncthreads();      // buffer s&1 consumed by all before next overwrite
  }

  // Block reduction -> one partial per block.
  red[tid] = acc0 + acc1;
  __syncthreads();
  for (int sft = BLOCK / 2; sft > 0; sft >>= 1) {
    if (tid < sft) red[tid] += red[tid + sft];
    __syncthreads();
  }
  if (tid == 0) ws[ib * GJ + jb] = red[0];
}

// ---------------- deterministic final reduction ----------------

__global__ __launch_bounds__(BLOCK) void
mse_bpr_reduce_kernel(const float* __restrict__ ws, float* __restrict__ out) {
  __shared__ float red[BLOCK];
  float v = 0.0f;
  for (int i = threadIdx.x; i < NBLOCKS; i += BLOCK) v += ws[i];
  red[threadIdx.x] = v;
  __syncthreads();
  for (int sft = BLOCK / 2; sft > 0; sft >>= 1) {
    if (threadIdx.x < sft) red[threadIdx.x] += red[threadIdx.x + sft];
    __syncthreads();
  }
  if (threadIdx.x == 0) {
    // accumulated (sp + mse_i) per masked pair == 2*x_ref; out = (2/N^2)*0.5*sum
    out[0] = red[0] * (1.0f / ((float)N_ELEMS * (float)N_ELEMS));
  }
}

// ---------------- launch ----------------

extern "C" void kernel_launch(void* const* d_in, const int* in_sizes, int n_in,
                              void* d_out, int out_size, void* d_ws, size_t ws_size,
                              hipStream_t stream) {
  (void)in_sizes; (void)n_in; (void)out_size; (void)ws_size;
  const float* x   = (const float*)d_in[0];   // "input"
  const float* t   = (const float*)d_in[1];   // "target"
  float*       out = (float*)d_out;
  float*       ws  = (float*)d_ws;            // NBLOCKS floats of scratch

  dim3 grid(GI, GJ);
  mse_bpr_pair_kernel<<<grid, BLOCK, 0, stream>>>(x, t, ws);
  mse_bpr_reduce_kernel<<<1, BLOCK, 0, stream>>>(ws, out);
}